// PositionalEncoding_87737591922760
// MI455X (gfx1250) — compile-verified
//
#include <hip/hip_runtime.h>
#include <stdint.h>

// Reference: out = pos-encoding [S=8192, D=2048] f32, broadcast over B=4.
// Input x is shape-only. Pure store-bandwidth problem: 256 MB -> ~11 us @ 23.3 TB/s.
#define PE_S 8192
#define PE_D 2048
#define PE_B 4
#define CHUNK 16   // rows per thread via sin/cos rotation recurrence

// -(log2(10000) / 1024): w_i = 10000^(-i/1024) = exp2(i * NEG_SCALE)
#define NEG_SCALE (-0.012976281620653759f)
// 10000^(-1/1024): ratio between consecutive inverse-frequencies
#define FREQ_RATIO 0.9910458559f

// One PE step: drain-to-4 async wait, stage float4 in LDS slot, wait DS,
// fire 4 async LDS->global stores (one per batch image, 64 MiB apart) at
// compile-time row offset OFFSTR, then rotate both (sin,cos) pairs by w.
#define PE_STEP(SLOT, OFFSTR)                                                  \
  do {                                                                         \
    asm volatile("s_wait_asynccnt 4" ::: "memory");                            \
    tile[SLOT][tid] = make_float4(s0, c0, s1, c1);                             \
    asm volatile(                                                              \
        "s_wait_dscnt 0\n\t"                                                   \
        "global_store_async_from_lds_b128 %0, %4, off offset:" OFFSTR "\n\t"   \
        "global_store_async_from_lds_b128 %1, %4, off offset:" OFFSTR "\n\t"   \
        "global_store_async_from_lds_b128 %2, %4, off offset:" OFFSTR "\n\t"   \
        "global_store_async_from_lds_b128 %3, %4, off offset:" OFFSTR          \
        :                                                                      \
        : "v"(g0), "v"(g1), "v"(g2), "v"(g3), "v"(lds_off##SLOT)               \
        : "memory");                                                           \
    float ns0 = s0 * cw0 + c0 * sw0;                                           \
    float nc0 = c0 * cw0 - s0 * sw0;                                           \
    float ns1 = s1 * cw1 + c1 * sw1;                                           \
    float nc1 = c1 * cw1 - s1 * sw1;                                           \
    s0 = ns0; c0 = nc0; s1 = ns1; c1 = nc1;                                    \
  } while (0)

__global__ __launch_bounds__(256) void pe_bcast_kernel(float* __restrict__ out) {
    __shared__ float4 tile[2][256];   // double-buffered 16B slot per lane (8 KB)

    const unsigned tid = threadIdx.x;
    const unsigned gid = blockIdx.x * 256u + tid;
    const unsigned q     = gid & 511u;          // column pair: floats [4q, 4q+3]
    const unsigned chunk = gid >> 9;            // row chunk in [0, 512)
    const unsigned p0    = chunk * CHUNK;       // first row of this thread

    // Inverse frequencies for i0 = 2q, i1 = 2q+1
    const float w0 = exp2f((float)(2u * q) * NEG_SCALE);
    const float w1 = w0 * FREQ_RATIO;
    const float fp0 = (float)p0;

    // Setup transcendentals (amortized over CHUNK rows):
    float s0, c0, s1, c1, sw0, cw0, sw1, cw1;
    sincosf(fp0 * w0, &s0, &c0);   // starting values at row p0
    sincosf(fp0 * w1, &s1, &c1);
    sincosf(w0, &sw0, &cw0);       // per-row rotation coefficients
    sincosf(w1, &sw1, &cw1);

    // Batch-0 byte address of (row p0, col 4q); batches are 64 MiB apart.
    const unsigned long long row_bytes =
        ((unsigned long long)p0 * (unsigned long long)PE_D + 4ull * q) * 4ull;
    unsigned long long g0 = (unsigned long long)(uintptr_t)out + row_bytes;
    const unsigned long long bstride =
        (unsigned long long)PE_S * (unsigned long long)PE_D * 4ull;
    unsigned long long g1 = g0 + bstride;
    unsigned long long g2 = g0 + 2ull * bstride;
    unsigned long long g3 = g0 + 3ull * bstride;

    // Workgroup-relative LDS byte offsets (low 32 bits of generic pointer)
    unsigned lds_off0 = (unsigned)(uintptr_t)&tile[0][tid];
    unsigned lds_off1 = (unsigned)(uintptr_t)&tile[1][tid];

    // 16 rows, row stride 8192 B encoded as the 24-bit immediate offset.
    PE_STEP(0, "0");
    PE_STEP(1, "8192");
    PE_STEP(0, "16384");
    PE_STEP(1, "24576");
    PE_STEP(0, "32768");
    PE_STEP(1, "40960");
    PE_STEP(0, "49152");
    PE_STEP(1, "57344");
    PE_STEP(0, "65536");
    PE_STEP(1, "73728");
    PE_STEP(0, "81920");
    PE_STEP(1, "90112");
    PE_STEP(0, "98304");
    PE_STEP(1, "106496");
    PE_STEP(0, "114688");
    PE_STEP(1, "122880");
    // s_endpgm performs an implicit wait-idle; outstanding ASYNCcnt is safe
    // and this block's LDS is never reused.
}

extern "C" void kernel_launch(void* const* d_in, const int* in_sizes, int n_in,
                              void* d_out, int out_size, void* d_ws, size_t ws_size,
                              hipStream_t stream) {
    (void)d_in; (void)in_sizes; (void)n_in; (void)d_ws; (void)ws_size; (void)out_size;
    float* out = (float*)d_out;

    // threads = (D/4 columns) * (S/CHUNK row-chunks) = 512 * 512 = 262144
    const unsigned total_threads = (PE_D / 4) * (PE_S / CHUNK);
    dim3 block(256);
    dim3 grid(total_threads / 256u);
    pe_bcast_kernel<<<grid, block, 0, stream>>>(out);
}